// SpatialChannelCrossAttention_78022375899285
// MI455X (gfx1250) — compile-verified
//
#include <hip/hip_runtime.h>

typedef __attribute__((ext_vector_type(2))) float v2f;
typedef __attribute__((ext_vector_type(8))) float v8f;

#define CCH   64
#define NSP   8192
#define NBAT  2
#define SCALE 0.125f   // 64^-0.5

__device__ __forceinline__ v8f vzero8() {
  v8f z = {0.f,0.f,0.f,0.f,0.f,0.f,0.f,0.f};
  return z;
}

// D = A(16x4 f32) * B(4x16 f32) + C  -- CDNA5 fp32 WMMA
__device__ __forceinline__ v8f wmma4(v2f a, v2f b, v8f c) {
  return __builtin_amdgcn_wmma_f32_16x16x4_f32(
      /*neg_a=*/false, a, /*neg_b=*/false, b,
      /*c_mod=*/(short)0, c, /*reuse_a=*/false, /*reuse_b=*/false);
}

// generic->LDS offset: flat->local addrspace cast == low-32-bit truncation
__device__ __forceinline__ uint32_t lds_off(const void* p) {
  return (uint32_t)(uintptr_t)p;
}

// CDNA5 async DMA: 16B global -> LDS, tracked by ASYNCcnt
__device__ __forceinline__ void async_g2l_b128(uint32_t dst_lds, const void* gptr) {
  asm volatile("global_load_async_to_lds_b128 %0, %1, off"
               :: "v"(dst_lds), "v"(gptr) : "memory");
}

// reductions across the 16 lanes of each wave32 half (xor masks stay in-half)
__device__ __forceinline__ float rmax16(float x) {
  x = fmaxf(x, __shfl_xor(x, 1));
  x = fmaxf(x, __shfl_xor(x, 2));
  x = fmaxf(x, __shfl_xor(x, 4));
  x = fmaxf(x, __shfl_xor(x, 8));
  return x;
}
__device__ __forceinline__ float rsum16(float x) {
  x += __shfl_xor(x, 1);
  x += __shfl_xor(x, 2);
  x += __shfl_xor(x, 4);
  x += __shfl_xor(x, 8);
  return x;
}

// ---------------------------------------------------------------------------
// Kernel 1: sf = w_s @ x + b_s ; cf = w_c @ x + b_c   (per batch, per voxel)
// grid = B * (N/64), block = 256 (8 waves)
// ---------------------------------------------------------------------------
__global__ __launch_bounds__(256) void proj_kernel(
    const float* __restrict__ x, const float* __restrict__ w_s,
    const float* __restrict__ b_s, const float* __restrict__ w_c,
    const float* __restrict__ b_c, float* __restrict__ sf,
    float* __restrict__ cf)
{
  __shared__ float xs [64 * 66];
  __shared__ float wS [64 * 66];
  __shared__ float wC [64 * 66];
  const int tid  = threadIdx.x;
  const int b    = blockIdx.x >> 7;
  const int nb   = (blockIdx.x & 127) << 6;
  for (int i = tid; i < 64 * 64; i += 256) {
    int c = i >> 6, j = i & 63;
    xs[c * 66 + j] = x[((size_t)((b << 6) + c)) * NSP + nb + j];
    wS[c * 66 + j] = w_s[(c << 6) + j];
    wC[c * 66 + j] = w_c[(c << 6) + j];
  }
  __syncthreads();
  const int lane = tid & 31, wv = tid >> 5;
  const int hl = lane >> 4, lr = lane & 15;
  const int nsub = (wv & 3) << 4;
  const int opr  = (wv >> 2) << 1;
  for (int t = 0; t < 2; ++t) {
    const int ob = (opr + t) << 4;
    v8f accS = vzero8(), accC = vzero8();
    for (int kb = 0; kb < 64; kb += 4) {
      const int k0 = kb + (hl << 1);
      v2f bx; bx.x = xs[k0 * 66 + nsub + lr]; bx.y = xs[(k0 + 1) * 66 + nsub + lr];
      v2f aS; aS.x = wS[(ob + lr) * 66 + k0]; aS.y = wS[(ob + lr) * 66 + k0 + 1];
      v2f aC; aC.x = wC[(ob + lr) * 66 + k0]; aC.y = wC[(ob + lr) * 66 + k0 + 1];
      accS = wmma4(aS, bx, accS);
      accC = wmma4(aC, bx, accC);
    }
#pragma unroll
    for (int v = 0; v < 8; ++v) {
      const int o = ob + v + (hl << 3);
      const size_t idx = ((size_t)((b << 6) + o)) * NSP + nb + nsub + lr;
      sf[idx] = accS[v] + b_s[o];
      cf[idx] = accC[v] + b_c[o];
    }
  }
}

// ---------------------------------------------------------------------------
// Kernel 2: s2c logits[b,c,k] = scale * sum_n sf[c,n]*cf[k,n]   (64x64, K=8192)
// grid = B, block = 256
// ---------------------------------------------------------------------------
__global__ __launch_bounds__(256) void s2c_logits_kernel(
    const float* __restrict__ sf, const float* __restrict__ cf,
    float* __restrict__ lg)
{
  __shared__ float sfs[64 * 66];
  __shared__ float cfs[64 * 66];
  const int tid = threadIdx.x;
  const int b   = blockIdx.x;
  const int lane = tid & 31, wv = tid >> 5;
  const int hl = lane >> 4, lr = lane & 15;
  const int t0 = wv << 1, t1 = t0 + 1;
  const int ct0 = t0 >> 2, kt0 = t0 & 3;
  const int ct1 = t1 >> 2, kt1 = t1 & 3;
  v8f acc0 = vzero8(), acc1 = vzero8();
  for (int nb = 0; nb < NSP; nb += 64) {
    __syncthreads();
    for (int i = tid; i < 64 * 64; i += 256) {
      int c = i >> 6, j = i & 63;
      const size_t g = ((size_t)((b << 6) + c)) * NSP + nb + j;
      sfs[c * 66 + j] = sf[g];
      cfs[c * 66 + j] = cf[g];
    }
    __syncthreads();
    for (int kb = 0; kb < 64; kb += 4) {
      const int k0 = kb + (hl << 1);
      v2f a0; a0.x = sfs[((ct0 << 4) + lr) * 66 + k0]; a0.y = sfs[((ct0 << 4) + lr) * 66 + k0 + 1];
      v2f b0; b0.x = cfs[((kt0 << 4) + lr) * 66 + k0]; b0.y = cfs[((kt0 << 4) + lr) * 66 + k0 + 1];
      acc0 = wmma4(a0, b0, acc0);
      v2f a1; a1.x = sfs[((ct1 << 4) + lr) * 66 + k0]; a1.y = sfs[((ct1 << 4) + lr) * 66 + k0 + 1];
      v2f b1; b1.x = cfs[((kt1 << 4) + lr) * 66 + k0]; b1.y = cfs[((kt1 << 4) + lr) * 66 + k0 + 1];
      acc1 = wmma4(a1, b1, acc1);
    }
  }
#pragma unroll
  for (int v = 0; v < 8; ++v) {
    const int r = v + (hl << 3);
    lg[b * 4096 + ((ct0 << 4) + r) * 64 + (kt0 << 4) + lr] = acc0[v] * SCALE;
    lg[b * 4096 + ((ct1 << 4) + r) * 64 + (kt1 << 4) + lr] = acc1[v] * SCALE;
  }
}

// ---------------------------------------------------------------------------
// Kernel 3: softmax over k of the 64x64 s2c logits. grid = B, block = 64
// ---------------------------------------------------------------------------
__global__ void s2c_softmax_kernel(const float* __restrict__ lg,
                                   float* __restrict__ attn)
{
  const int b = blockIdx.x, r = threadIdx.x;
  const float* row = lg + b * 4096 + r * 64;
  float m = -3.0e38f;
  for (int k = 0; k < 64; ++k) m = fmaxf(m, row[k]);
  float s = 0.f;
  for (int k = 0; k < 64; ++k) s += __expf(row[k] - m);
  const float rs = 1.0f / s;
  for (int k = 0; k < 64; ++k)
    attn[b * 4096 + r * 64 + k] = __expf(row[k] - m) * rs;
}

// ---------------------------------------------------------------------------
// Kernel 4: c2s flash pass 1 — per-row (n) online max M and 1/Z over all m.
// l[n,m] = scale * sum_c cf[c,n]*sf[c,m].  grid = B*(N/128), block = 256
// sf m-chunks (32 cols) are double-buffered via async global->LDS DMA.
// ---------------------------------------------------------------------------
__global__ __launch_bounds__(256) void c2s_stats_kernel(
    const float* __restrict__ sf, const float* __restrict__ cf,
    float* __restrict__ Mbuf, float* __restrict__ Rbuf)
{
  __shared__ float cfs[64 * 130];    // cf n-block (A operand, resident)
  __shared__ float sfs[2][64 * 36];  // sf m-chunks (B operand, async dbl-buf)
  const int tid  = threadIdx.x;
  const int b    = blockIdx.x >> 6;
  const int nblk = (blockIdx.x & 63) << 7;
  const size_t sbase = ((size_t)b << 6) * NSP;
  for (int i = tid; i < 64 * 128; i += 256) {
    int c = i >> 7, j = i & 127;
    cfs[c * 130 + j] = cf[sbase + (size_t)c * NSP + nblk + j];
  }
  // per-thread async share: 2x b128 per 32-wide chunk
  const int ca = tid >> 3;            // rows 0..31 (first op) / +32 (second)
  const int qa = (tid & 7) << 2;      // float quad within 32-col chunk
  async_g2l_b128(lds_off(&sfs[0][ca * 36 + qa]),        sf + sbase + (size_t)ca * NSP + qa);
  async_g2l_b128(lds_off(&sfs[0][(ca + 32) * 36 + qa]), sf + sbase + (size_t)(ca + 32) * NSP + qa);

  const int lane = tid & 31, wv = tid >> 5;
  const int hl = lane >> 4, lr = lane & 15;
  float M[8], Z[8];
#pragma unroll
  for (int v = 0; v < 8; ++v) { M[v] = -3.0e38f; Z[v] = 0.f; }

  for (int mb = 0, it = 0; mb < NSP; mb += 32, ++it) {
    const int cur = it & 1, nxt = cur ^ 1;
    __syncthreads();  // nobody still reads buf[nxt]
    if (mb + 32 < NSP) {
      async_g2l_b128(lds_off(&sfs[nxt][ca * 36 + qa]),
                     sf + sbase + (size_t)ca * NSP + mb + 32 + qa);
      async_g2l_b128(lds_off(&sfs[nxt][(ca + 32) * 36 + qa]),
                     sf + sbase + (size_t)(ca + 32) * NSP + mb + 32 + qa);
      asm volatile("s_wait_asynccnt 2" ::: "memory");  // chunk `cur` retired
    } else {
      asm volatile("s_wait_asynccnt 0" ::: "memory");
    }
    __syncthreads();  // all waves' async writes visible
    const float* S = &sfs[cur][0];
    for (int ms = 0; ms < 2; ++ms) {
      v8f acc = vzero8();
      for (int kb = 0; kb < 64; kb += 4) {
        const int k0 = kb + (hl << 1);
        v2f a;  a.x  = cfs[k0 * 130 + (wv << 4) + lr]; a.y  = cfs[(k0 + 1) * 130 + (wv << 4) + lr];
        v2f bb; bb.x = S[k0 * 36 + (ms << 4) + lr];    bb.y = S[(k0 + 1) * 36 + (ms << 4) + lr];
        acc = wmma4(a, bb, acc);
      }
#pragma unroll
      for (int v = 0; v < 8; ++v) {
        const float xv = acc[v] * SCALE;
        const float tm = rmax16(xv);
        const float nm = fmaxf(M[v], tm);
        const float ts = rsum16(__expf(xv - nm));
        Z[v] = Z[v] * __expf(M[v] - nm) + ts;
        M[v] = nm;
      }
    }
  }
  if (lr == 0) {
#pragma unroll
    for (int v = 0; v < 8; ++v) {
      const int n = nblk + (wv << 4) + (hl << 3) + v;
      Mbuf[b * NSP + n] = M[v];
      Rbuf[b * NSP + n] = 1.0f / Z[v];   // store reciprocal (Z >= 1 always)
    }
  }
}

// ---------------------------------------------------------------------------
// Kernel 5: c2s flash pass 2 + s2c output + residual.
// out[c,m] = sum_n sf[c,n]*exp(l[n,m]-M[n])/Z[n] + sum_k attn[c,k]*cf[k,m] + x
// grid = B*(N/128) m-blocks, block = 256.
// cf/sf n-tiles double-buffered via async global->LDS DMA.
// ---------------------------------------------------------------------------
__global__ __launch_bounds__(256) void c2s_out_kernel(
    const float* __restrict__ x, const float* __restrict__ sf,
    const float* __restrict__ cf, const float* __restrict__ Mbuf,
    const float* __restrict__ Rbuf, const float* __restrict__ attn,
    float* __restrict__ out)
{
  __shared__ float sfs[64 * 130];    // sf m-block (B operand for logits)
  __shared__ float cft[2][64 * 20];  // cf n-tile  (A for logits, async dbl-buf)
  __shared__ float sft[2][64 * 20];  // sf n-tile  (A for out GEMM, async dbl-buf)
  __shared__ float Pst[8][16 * 18];  // per-wave probability tile
  const int tid  = threadIdx.x;
  const int b    = blockIdx.x >> 6;
  const int mblk = (blockIdx.x & 63) << 7;
  const size_t sbase = ((size_t)b << 6) * NSP;
  for (int i = tid; i < 64 * 128; i += 256) {
    int c = i >> 7, j = i & 127;
    sfs[c * 130 + j] = sf[sbase + (size_t)c * NSP + mblk + j];
  }
  // per-thread async share for a 64x16 tile pair: 1 b128 from cf + 1 from sf
  const int ca = tid >> 2;           // row 0..63
  const int qa = (tid & 3) << 2;     // float quad within 16-col tile
  async_g2l_b128(lds_off(&cft[0][ca * 20 + qa]), cf + sbase + (size_t)ca * NSP + qa);
  async_g2l_b128(lds_off(&sft[0][ca * 20 + qa]), sf + sbase + (size_t)ca * NSP + qa);

  const int lane = tid & 31, wv = tid >> 5;
  const int hl = lane >> 4, lr = lane & 15;
  const int msub = wv << 4;
  v8f accO[4];
#pragma unroll
  for (int ct = 0; ct < 4; ++ct) accO[ct] = vzero8();

  for (int nt = 0; nt < NSP / 16; ++nt) {
    const int cur = nt & 1, nxt = cur ^ 1;
    const int nb = nt << 4;
    __syncthreads();  // nobody still reads buf[nxt]
    if (nt + 1 < NSP / 16) {
      async_g2l_b128(lds_off(&cft[nxt][ca * 20 + qa]),
                     cf + sbase + (size_t)ca * NSP + nb + 16 + qa);
      async_g2l_b128(lds_off(&sft[nxt][ca * 20 + qa]),
                     sf + sbase + (size_t)ca * NSP + nb + 16 + qa);
      asm volatile("s_wait_asynccnt 2" ::: "memory");  // tile `cur` retired
    } else {
      asm volatile("s_wait_asynccnt 0" ::: "memory");
    }
    __syncthreads();  // all waves' async writes visible
    const float* CT = &cft[cur][0];
    const float* ST = &sft[cur][0];
    // logits tile l[16 n x 16 m], K = 64 channels
    v8f acc = vzero8();
    for (int kb = 0; kb < 64; kb += 4) {
      const int k0 = kb + (hl << 1);
      v2f a;  a.x  = CT[k0 * 20 + lr];           a.y  = CT[(k0 + 1) * 20 + lr];
      v2f bb; bb.x = sfs[k0 * 130 + msub + lr];  bb.y = sfs[(k0 + 1) * 130 + msub + lr];
      acc = wmma4(a, bb, acc);
    }
    // P = exp(scale*l - M[n]) * (1/Z[n]); stage through per-wave LDS to
    // re-fragment D-layout -> B-layout for the next GEMM.
    float* Pw = &Pst[wv][0];
#pragma unroll
    for (int v = 0; v < 8; ++v) {
      const int n = nb + (hl << 3) + v;
      const float p = __expf(acc[v] * SCALE - Mbuf[b * NSP + n]) * Rbuf[b * NSP + n];
      Pw[((hl << 3) + v) * 18 + lr] = p;
    }
    asm volatile("s_wait_dscnt 0" ::: "memory");  // wave-local LDS RAW fence
    // accO[c-tile] += sf[c, n-tile] @ P[n-tile, m]
#pragma unroll
    for (int ct = 0; ct < 4; ++ct) {
      for (int kb = 0; kb < 16; kb += 4) {
        const int k0 = kb + (hl << 1);
        v2f a;  a.x  = ST[((ct << 4) + lr) * 20 + k0]; a.y  = ST[((ct << 4) + lr) * 20 + k0 + 1];
        v2f bb; bb.x = Pw[k0 * 18 + lr];               bb.y = Pw[(k0 + 1) * 18 + lr];
        accO[ct] = wmma4(a, bb, accO[ct]);
      }
    }
  }
  // s2c output: accO += attn[c,k] @ cf[k,m]   (K = 64, tiny)
#pragma unroll
  for (int ct = 0; ct < 4; ++ct) {
    for (int kb = 0; kb < 64; kb += 4) {
      const int k0 = kb + (hl << 1);
      v2f a;
      a.x = attn[b * 4096 + ((ct << 4) + lr) * 64 + k0];
      a.y = attn[b * 4096 + ((ct << 4) + lr) * 64 + k0 + 1];
      const size_t g0 = sbase + (size_t)k0 * NSP + mblk + msub + lr;
      v2f bb; bb.x = cf[g0]; bb.y = cf[g0 + NSP];
      accO[ct] = wmma4(a, bb, accO[ct]);
    }
  }
  // residual + store
#pragma unroll
  for (int ct = 0; ct < 4; ++ct) {
#pragma unroll
    for (int v = 0; v < 8; ++v) {
      const int c = (ct << 4) + (hl << 3) + v;
      const size_t idx = sbase + (size_t)c * NSP + mblk + msub + lr;
      out[idx] = accO[ct][v] + x[idx];
    }
  }
}

// ---------------------------------------------------------------------------
extern "C" void kernel_launch(void* const* d_in, const int* in_sizes, int n_in,
                              void* d_out, int out_size, void* d_ws, size_t ws_size,
                              hipStream_t stream)
{
  const float* x   = (const float*)d_in[0];
  const float* w_s = (const float*)d_in[1];
  const float* b_s = (const float*)d_in[2];
  const float* w_c = (const float*)d_in[3];
  const float* b_c = (const float*)d_in[4];
  float* out = (float*)d_out;

  float* ws = (float*)d_ws;
  float* sf = ws;                                   // [B,64,N]
  float* cf = sf + (size_t)NBAT * CCH * NSP;        // [B,64,N]
  float* Mb = cf + (size_t)NBAT * CCH * NSP;        // [B,N] row max
  float* Rb = Mb + (size_t)NBAT * NSP;              // [B,N] 1/rowsum
  float* lg = Rb + (size_t)NBAT * NSP;              // [B,64,64]
  float* at = lg + NBAT * CCH * CCH;                // [B,64,64]

  proj_kernel       <<<NBAT * (NSP / 64), 256, 0, stream>>>(x, w_s, b_s, w_c, b_c, sf, cf);
  s2c_logits_kernel <<<NBAT,              256, 0, stream>>>(sf, cf, lg);
  s2c_softmax_kernel<<<NBAT,               64, 0, stream>>>(lg, at);
  c2s_stats_kernel  <<<NBAT * (NSP / 128), 256, 0, stream>>>(sf, cf, Mb, Rb);
  c2s_out_kernel    <<<NBAT * (NSP / 128), 256, 0, stream>>>(x, sf, cf, Mb, Rb, at, out);
}